// PSDShift_46840913330850
// MI455X (gfx1250) — compile-verified
//
#include <hip/hip_runtime.h>
#include <math.h>

typedef __attribute__((ext_vector_type(2))) float v2f;
typedef __attribute__((ext_vector_type(8))) float v8f;

#define NSEG   800
#define NBATCH 256
#define NPER   100
#define NFREQ  51
#define NCOL   102          // 2*NFREQ (cos/sin interleaved)
#define NCOLP  112          // padded to 7 tiles of 16 columns
#define KSTR   101          // LDS k-stride (odd -> no bank conflicts)
#define FLATN  13056        // NBATCH*NFREQ
#define MINV   1e-6f
#define TWOPI  6.28318530717958647692f

// ---------------------------------------------------------------------------
// Kernel 1: windowed-DFT as GEMM via V_WMMA_F32_16X16X4_F32, fused PSD + log.
// X viewed as [204800, 100]; W^T staged in LDS. Output P[seg][b*51+f].
// B fragments for all 7 N-tiles are preloaded per k-step so the 7 ds_loads
// are issued back-to-back and the 7 WMMAs execute without per-op dscnt waits.
// ---------------------------------------------------------------------------
__global__ __launch_bounds__(256)
void psd_gemm_kernel(const float* __restrict__ X, float* __restrict__ P) {
    __shared__ float Wt[NCOLP * KSTR];   // Wt[c*KSTR + k], ~45 KB

    // Build transposed, window-folded DFT basis. Padded cols (>=102) are zero.
    for (int idx = threadIdx.x; idx < NCOLP * KSTR; idx += 256) {
        int c = idx / KSTR;
        int k = idx % KSTR;
        float v = 0.f;
        if (c < NCOL && k < NPER) {
            float kf  = (float)k;
            float win = 1.f - cosf(TWOPI * kf * (1.0f / 100.0f)); // hann*2
            int   f   = c >> 1;
            float ang = TWOPI * kf * (float)f * (1.0f / 100.0f);
            v = (c & 1) ? (-win * sinf(ang)) : (win * cosf(ang));
        }
        Wt[idx] = v;
    }
    __syncthreads();

    const int wave = threadIdx.x >> 5;
    const int lane = threadIdx.x & 31;
    const int m    = lane & 15;
    const int koff = (lane < 16) ? 0 : 2;
    const int s0   = (blockIdx.x * 8 + wave) * 16;   // 16 segment-rows per wave

    const float* rowPtr = X + (size_t)(s0 + m) * NPER;
    const float* wbase  = &Wt[m * KSTR + koff];

    v8f acc[7] = {};

    for (int kk = 0; kk < NPER; kk += 4) {
        // A fragment: 16x4 f32, lanes 0-15 hold K=kk..kk+1, lanes 16-31 K=kk+2..kk+3
        v2f a;
        a.x = fmaxf(rowPtr[kk + koff],     MINV);
        a.y = fmaxf(rowPtr[kk + koff + 1], MINV);

        // Preload all 7 B fragments (independent LDS loads, one wait).
        v2f b[7];
        #pragma unroll
        for (int n = 0; n < 7; ++n) {
            const float* wp = wbase + (16 * n) * KSTR + kk;
            b[n].x = wp[0];
            b[n].y = wp[1];
        }
        #pragma unroll
        for (int n = 0; n < 7; ++n) {
            acc[n] = __builtin_amdgcn_wmma_f32_16x16x4_f32(
                false, a, false, b[n], (short)0, acc[n], false, false);
        }
    }

    // Epilogue: psd = cos^2 + sin^2 (adjacent columns -> lane-pair shfl),
    // then signed clamped log, scatter to P[seg][b*51+f].
    const int rbase = s0 + ((lane < 16) ? 0 : 8);
    #pragma unroll
    for (int n = 0; n < 7; ++n) {
        int  f      = 8 * n + (m >> 1);
        bool writer = ((m & 1) == 0) && (f < NFREQ);
        #pragma unroll
        for (int vg = 0; vg < 8; ++vg) {
            float v  = acc[n][vg];
            float sq = v * v;
            float p  = sq + __shfl_xor(sq, 1, 32);   // convergent: all lanes
            if (writer) {
                int   srow = rbase + vg;
                int   seg  = srow % NSEG;
                int   bb   = srow / NSEG;
                float lx   = logf(p);
                float sgn  = (lx > 0.f) ? 1.f : ((lx < 0.f) ? -1.f : 0.f);
                P[(size_t)seg * FLATN + bb * NFREQ + f] =
                    sgn * fmaxf(fabsf(lx), MINV);
            }
        }
    }
}

// ---------------------------------------------------------------------------
// Kernel 2: faithful flat-view normalization. Block = one segment; thread g
// owns the 51-value group {g + 256*t}. Values held in registers.
// ---------------------------------------------------------------------------
__global__ __launch_bounds__(256)
void normalize_kernel(float* __restrict__ P) {
    float* base = P + (size_t)blockIdx.x * FLATN;
    const int g = threadIdx.x;
    float vals[NFREQ];
    float mn =  3.4e38f, mx = -3.4e38f;
    #pragma unroll
    for (int t = 0; t < NFREQ; ++t) {
        float v = base[t * 256 + g];
        vals[t] = v;
        mn = fminf(mn, v);
        mx = fmaxf(mx, v);
    }
    float inv = 1.f / (mx - mn);   // reference has no guard either
    #pragma unroll
    for (int t = 0; t < NFREQ; ++t)
        base[t * 256 + g] = (vals[t] - mn) * inv;
}

// ---------------------------------------------------------------------------
// Kernel 3: column sums over segments: u[k] = sum_seg y, q[k] = sum_seg y^2.
// ---------------------------------------------------------------------------
__global__ __launch_bounds__(256)
void colsum_kernel(const float* __restrict__ P,
                   float* __restrict__ u, float* __restrict__ q) {
    int k = blockIdx.x * 256 + threadIdx.x;
    if (k >= FLATN) return;
    float su = 0.f, sq = 0.f;
    for (int seg = 0; seg < NSEG; ++seg) {
        float v = P[(size_t)seg * FLATN + k];
        su += v;
        sq += v * v;
    }
    u[k] = su;
    q[k] = sq;
}

// ---------------------------------------------------------------------------
// Kernel 4: final rank-1 reductions -> the two scalar losses.
//  psd_loss  = (N*SF + N*SR - 2*u_f.u_r) / (N*N*n)
//  psd_self  = (N*SF - |u_f|^2) / (n * N*(N-1)/2)
// ---------------------------------------------------------------------------
__global__ __launch_bounds__(256)
void final_kernel(const float* __restrict__ uf, const float* __restrict__ qf,
                  const float* __restrict__ ur, const float* __restrict__ qr,
                  float* __restrict__ out) {
    __shared__ float red[4][256];
    float SF = 0.f, SR = 0.f, UU = 0.f, UV = 0.f;
    for (int k = threadIdx.x; k < FLATN; k += 256) {
        float a = uf[k], b = ur[k];
        SF += qf[k];
        SR += qr[k];
        UU += a * a;
        UV += a * b;
    }
    int t = threadIdx.x;
    red[0][t] = SF; red[1][t] = SR; red[2][t] = UU; red[3][t] = UV;
    __syncthreads();
    for (int s = 128; s > 0; s >>= 1) {
        if (t < s) {
            red[0][t] += red[0][t + s];
            red[1][t] += red[1][t + s];
            red[2][t] += red[2][t + s];
            red[3][t] += red[3][t + s];
        }
        __syncthreads();
    }
    if (t == 0) {
        float sf = red[0][0], sr = red[1][0], uu = red[2][0], uv = red[3][0];
        const float N = 800.f, n = (float)FLATN;
        const float m = 800.f * 799.f * 0.5f;   // 319600
        out[0] = (N * sf + N * sr - 2.f * uv) / (N * N * n);
        out[1] = (N * sf - uu) / (n * m);
    }
}

// ---------------------------------------------------------------------------
extern "C" void kernel_launch(void* const* d_in, const int* in_sizes, int n_in,
                              void* d_out, int out_size, void* d_ws, size_t ws_size,
                              hipStream_t stream) {
    const float* fake = (const float*)d_in[0];
    const float* real = (const float*)d_in[1];
    float* out = (float*)d_out;

    float* ws = (float*)d_ws;
    float* P  = ws;                                   // 800*13056 floats (~42 MB), reused
    float* uf = ws + (size_t)NSEG * FLATN;
    float* qf = uf + FLATN;
    float* ur = qf + FLATN;
    float* qr = ur + FLATN;

    const int gemmBlocks = (NBATCH * NSEG) / 128;     // 1600 (8 waves x 16 rows)

    // fake
    psd_gemm_kernel<<<gemmBlocks, 256, 0, stream>>>(fake, P);
    normalize_kernel<<<NSEG, 256, 0, stream>>>(P);
    colsum_kernel<<<FLATN / 256, 256, 0, stream>>>(P, uf, qf);
    // real (reuse P)
    psd_gemm_kernel<<<gemmBlocks, 256, 0, stream>>>(real, P);
    normalize_kernel<<<NSEG, 256, 0, stream>>>(P);
    colsum_kernel<<<FLATN / 256, 256, 0, stream>>>(P, ur, qr);

    final_kernel<<<1, 256, 0, stream>>>(uf, qf, ur, qr, out);
}